// Spatial_Edge_Enhance_63513976373866
// MI455X (gfx1250) — compile-verified
//
#include <hip/hip_runtime.h>

typedef __attribute__((ext_vector_type(2))) float v2f;
typedef __attribute__((ext_vector_type(4))) float v4f;
typedef __attribute__((ext_vector_type(8))) float v8f;

#define Bn 2048
#define Jn 22
#define Dn 128
#define SS 132   // padded LDS row stride (floats): bank-conflict-free, 16B aligned
#define WS 132
#define YS 132

__global__ __launch_bounds__(256) void spatial_edge_enhance_kernel(
    const float* __restrict__ src,   // [B, J, D]
    const float* __restrict__ W,     // [D, D]  row-major [e][d]
    const float* __restrict__ bias,  // [D]
    float* __restrict__ out)         // [B, J, J, D]
{
    __shared__ __align__(16) float srcLDS[32 * SS];   // J padded 22->32
    __shared__ __align__(16) float wLDS[128 * WS];
    __shared__ __align__(16) float yLDS[32 * YS];
    __shared__ __align__(16) float bLDS[Dn];

    const int b   = blockIdx.x;
    const int tid = threadIdx.x;

    // ---- Phase 0: cooperative loads into LDS ----
    // Zero the padding rows 22..31 of the src tile (so padded-M WMMA adds 0).
    for (int t = tid; t < (32 - Jn) * SS; t += 256)
        srcLDS[Jn * SS + t] = 0.0f;

    // src[b] : 22*128 floats = 704 float4s, coalesced global reads
    const float* sb = src + (size_t)b * (Jn * Dn);
    for (int q = tid; q < (Jn * Dn) / 4; q += 256) {
        int j = q >> 5, dq = q & 31;
        *(v4f*)&srcLDS[j * SS + dq * 4] = *(const v4f*)(sb + q * 4);
    }
    // W : 128*128 floats = 4096 float4s (L2-resident across all 2048 blocks)
    for (int q = tid; q < (Dn * Dn) / 4; q += 256) {
        int e = q >> 5, dq = q & 31;
        *(v4f*)&wLDS[e * WS + dq * 4] = *(const v4f*)(W + q * 4);
    }
    if (tid < Dn / 4)
        *(v4f*)&bLDS[tid * 4] = *(const v4f*)(bias + tid * 4);
    __syncthreads();

    // ---- Phase 1: Y[b] = src[b] @ W^T via V_WMMA_F32_16X16X4_F32 ----
    // Per wave: N-tile e0 = wave*16; both M-tiles (rows 0-15, 16-31pad); K=128 in 32 steps.
    const int lane = tid & 31;          // wave32
    const int wave = tid >> 5;          // 8 waves
    const int lm   = lane & 15;
    const int klo  = (lane >> 4) << 1;  // 0 for lanes 0-15, 2 for lanes 16-31
    const int e0   = wave << 4;

    for (int mt = 0; mt < 2; ++mt) {
        const int m0 = mt << 4;
        v8f acc = {};
        // A[m][k] = srcLDS[m0+lm][k0+klo..+1]   (16x4 f32 A layout)
        // B[k][n] = W[e0+lm][k0+klo..+1]        (4x16 f32 B layout; B = W^T)
        const float* arow = &srcLDS[(m0 + lm) * SS + klo];
        const float* brow = &wLDS[(e0 + lm) * WS + klo];
#pragma unroll
        for (int kk = 0; kk < 32; ++kk) {
            v2f a  = *(const v2f*)(arow + (kk << 2));
            v2f bm = *(const v2f*)(brow + (kk << 2));
            // 8 args: (neg_a, A, neg_b, B, c_mod, C, reuse_a, reuse_b)
            acc = __builtin_amdgcn_wmma_f32_16x16x4_f32(
                false, a, false, bm, (short)0, acc, false, false);
        }
        // C/D layout: VGPR r -> row m0+r (lanes 0-15) / m0+r+8 (lanes 16-31), col e0+lm
        const int radd = (lane < 16) ? 0 : 8;
#pragma unroll
        for (int r = 0; r < 8; ++r)
            yLDS[(m0 + r + radd) * YS + e0 + lm] = acc[r];
    }
    __syncthreads();

    // ---- Phase 2: stream out[b,i,j,:] = Y[j] - Y[i] + bias  (store-BW bound) ----
    // Flat t over 22*22*32 float4s; output float offset == t*4; each wave of 32
    // lanes covers one (i,j) row -> fully coalesced 512B non-temporal stores.
    float* ob = out + (size_t)b * (Jn * Jn * Dn);
    const int TOT = Jn * Jn * (Dn / 4);  // 15488
    for (int t = tid; t < TOT; t += 256) {
        int i = t / (Jn * (Dn / 4));     // t / 704
        int r = t - i * (Jn * (Dn / 4));
        int j = r >> 5;
        int q = (r & 31) << 2;
        v4f yj = *(const v4f*)&yLDS[j * YS + q];
        v4f yi = *(const v4f*)&yLDS[i * YS + q];
        v4f bb = *(const v4f*)&bLDS[q];
        v4f val = yj - yi + bb;
        __builtin_nontemporal_store(val, (v4f*)(ob + (size_t)t * 4));
    }
}

extern "C" void kernel_launch(void* const* d_in, const int* in_sizes, int n_in,
                              void* d_out, int out_size, void* d_ws, size_t ws_size,
                              hipStream_t stream) {
    const float* src  = (const float*)d_in[0];  // [2048, 22, 128] f32
    const float* W    = (const float*)d_in[1];  // [128, 128] f32
    const float* bias = (const float*)d_in[2];  // [128] f32
    float* out = (float*)d_out;                 // [2048, 22, 22, 128] f32
    spatial_edge_enhance_kernel<<<Bn, 256, 0, stream>>>(src, W, bias, out);
}